// NotearsRKHS_82446192214185
// MI455X (gfx1250) — compile-verified
//
#include <hip/hip_runtime.h>

// NotearsRKHS on gfx1250: per-l 16x24x24 FP32 GEMM via V_WMMA_F32_16X16X4_F32,
// exp-weighted combine on VALU/TRANS pipe directly in the WMMA C/D layout.

typedef __attribute__((ext_vector_type(2))) float v2f;
typedef __attribute__((ext_vector_type(8))) float v8f;

#define NPTS 2000
#define DDIM 24
#define NTILE 125              // 125 * 16 == 2000 exactly
#define LSPLIT 20
#define LCHUNK (NPTS / LSPLIT) // 100

#define BETAT_FLOATS  (NPTS * DDIM * DDIM)   // 1,152,000
#define ALPHAT_FLOATS (NPTS * DDIM)          // 48,000
#define PART_FLOATS   (LSPLIT * NPTS * DDIM) // 960,000

// ---------------------------------------------------------------------------
// Kernel 1: betaT[l][j][a] = (a==j ? 0 : beta[j][a][l]); alphaT[l][j] = alpha[j][l]
// ---------------------------------------------------------------------------
__global__ __launch_bounds__(256) void prep_kernel(const float* __restrict__ alpha,
                                                   const float* __restrict__ beta,
                                                   float* __restrict__ betaT,
                                                   float* __restrict__ alphaT) {
  int t = blockIdx.x * blockDim.x + threadIdx.x;
  if (t < BETAT_FLOATS) {
    int l = t / (DDIM * DDIM);
    int rem = t - l * (DDIM * DDIM);
    int j = rem / DDIM;
    int a = rem - j * DDIM;
    betaT[t] = (a == j) ? 0.0f : beta[(j * DDIM + a) * NPTS + l];
  }
  if (t < ALPHAT_FLOATS) {
    int l = t / DDIM;
    int j = t - l * DDIM;
    alphaT[t] = alpha[j * NPTS + l];
  }
}

// ---------------------------------------------------------------------------
// Kernel 2: main. One wave32 per block. blockIdx.x = i-tile (16 rows),
// blockIdx.y = l-chunk. Accumulators live in the WMMA 16x16 f32 C/D layout:
//   column n = lane&15, rows rowbase..rowbase+7 with rowbase = (lane>>4)*8.
// ---------------------------------------------------------------------------
__global__ __launch_bounds__(32) void rkhs_main(const float* __restrict__ x,
                                                const float* __restrict__ alphaT,
                                                const float* __restrict__ betaT,
                                                float* __restrict__ partial) {
  const int itile = blockIdx.x;
  const int chunk = blockIdx.y;
  const int lane  = threadIdx.x;
  const int i0    = itile * 16;

  __shared__ float xs[16][DDIM];       // x tile (16 rows)
  __shared__ float xl[DDIM];           // x[l,:]
  __shared__ float df[16][DDIM + 2];   // diff tile, padded rows
  __shared__ float Ssum[16];           // S[i,l] per tile row

  for (int t = lane; t < 16 * DDIM; t += 32) {
    int r = t / DDIM, k = t - r * DDIM;
    xs[r][k] = x[(i0 + r) * DDIM + k];
  }
  __syncthreads();

  const int  n0      = lane & 15;          // output column within j-tile
  const int  koff    = (lane >> 4) * 2;    // A/B fragment: half-wave K offset
  const int  rowbase = (lane >> 4) * 8;    // C/D fragment: half-wave row base
  const bool jt1ok   = (n0 < 8);           // second j-tile covers j=16..23 only
  const int  j1      = n0;
  const int  j2      = jt1ok ? (16 + n0) : 0; // clamped safe index

  float acc0[8], acc1[8];
#pragma unroll
  for (int m = 0; m < 8; ++m) { acc0[m] = 0.0f; acc1[m] = 0.0f; }

  const int l0 = chunk * LCHUNK;
  for (int l = l0; l < l0 + LCHUNK; ++l) {
    if (lane < DDIM) xl[lane] = x[l * DDIM + lane];
    __syncthreads();

    for (int t = lane; t < 16 * DDIM; t += 32) {
      int r = t / DDIM, k = t - r * DDIM;
      df[r][k] = xs[r][k] - xl[k];
    }
    __syncthreads();

    if (lane < 16) {
      float s = 0.0f;
#pragma unroll
      for (int k = 0; k < DDIM; ++k) s += df[lane][k] * df[lane][k];
      Ssum[lane] = s;
    }
    __syncthreads();

    // G[r][j] = sum_a df[r][a] * betaT[l][j][a], via 6 K-steps x 2 j-tiles of
    // V_WMMA_F32_16X16X4_F32. EXEC is full here (straight-line, single wave).
    const float* Bl = betaT + (size_t)l * (DDIM * DDIM);
    v8f c0 = {};
    v8f c1 = {};
#pragma unroll
    for (int k0 = 0; k0 < DDIM; k0 += 4) {
      // A fragment (16x4 f32): lane&15 = M row; VGPR0/1 hold K=k0+koff, +1.
      v2f a;
      a.x = df[n0][k0 + koff];
      a.y = df[n0][k0 + koff + 1];
      // B fragment (4x16 f32), tile 0: column j1.
      v2f b0;
      b0.x = Bl[j1 * DDIM + k0 + koff];
      b0.y = Bl[j1 * DDIM + k0 + koff + 1];
      // Tile 1: column j2 (zero-padded for j >= 24).
      float t0 = Bl[j2 * DDIM + k0 + koff];
      float t1 = Bl[j2 * DDIM + k0 + koff + 1];
      v2f b1;
      b1.x = jt1ok ? t0 : 0.0f;
      b1.y = jt1ok ? t1 : 0.0f;
      c0 = __builtin_amdgcn_wmma_f32_16x16x4_f32(false, a, false, b0,
                                                 (short)0, c0, false, false);
      c1 = __builtin_amdgcn_wmma_f32_16x16x4_f32(false, a, false, b1,
                                                 (short)0, c1, false, false);
    }

    // Combine: out[i,j] += exp(d_j^2 - S) * (alpha[j,l] + 2*G[i,j])
    float al0  = alphaT[l * DDIM + j1];
    float al1t = alphaT[l * DDIM + j2];
    float al1  = jt1ok ? al1t : 0.0f;
#pragma unroll
    for (int m = 0; m < 8; ++m) {
      int   r  = rowbase + m;
      float S  = Ssum[r];
      float d0 = df[r][j1];
      float w0 = __expf(d0 * d0 - S);
      acc0[m] += w0 * (al0 + 2.0f * c0[m]);
      float d1 = df[r][j2];
      float w1 = __expf(d1 * d1 - S);
      acc1[m] += w1 * (al1 + 2.0f * c1[m]);   // al1==0 && c1==0 for pad lanes
    }
    __syncthreads();   // df/xl reused next iteration
  }

  float* p = partial + ((size_t)chunk * NPTS + i0) * DDIM;
#pragma unroll
  for (int m = 0; m < 8; ++m) {
    int r = rowbase + m;
    p[r * DDIM + j1] = acc0[m];
    if (jt1ok) p[r * DDIM + 16 + n0] = acc1[m];
  }
}

// ---------------------------------------------------------------------------
// Kernel 3: deterministic reduction over l-chunks.
// ---------------------------------------------------------------------------
__global__ __launch_bounds__(256) void reduce_kernel(const float* __restrict__ partial,
                                                     float* __restrict__ out) {
  int t = blockIdx.x * blockDim.x + threadIdx.x;
  if (t < NPTS * DDIM) {
    float s = 0.0f;
#pragma unroll
    for (int c = 0; c < LSPLIT; ++c)
      s += partial[(size_t)c * (NPTS * DDIM) + t];
    out[t] = s;
  }
}

extern "C" void kernel_launch(void* const* d_in, const int* in_sizes, int n_in,
                              void* d_out, int out_size, void* d_ws, size_t ws_size,
                              hipStream_t stream) {
  const float* x     = (const float*)d_in[0];  // [N, D]
  const float* alpha = (const float*)d_in[1];  // [D, N]
  const float* beta  = (const float*)d_in[2];  // [D, D, N]
  float* out = (float*)d_out;                  // [N, D] f32

  char* ws = (char*)d_ws;
  float* betaT   = (float*)ws;                                            // 4.608 MB
  float* alphaT  = (float*)(ws + (size_t)BETAT_FLOATS * sizeof(float));   // 192 KB
  float* partial = (float*)(ws + (size_t)(BETAT_FLOATS + ALPHAT_FLOATS) * sizeof(float)); // 3.84 MB

  (void)in_sizes; (void)n_in; (void)out_size; (void)ws_size;

  prep_kernel<<<(BETAT_FLOATS + 255) / 256, 256, 0, stream>>>(alpha, beta, betaT, alphaT);

  dim3 grid(NTILE, LSPLIT);
  rkhs_main<<<grid, 32, 0, stream>>>(x, alphaT, betaT, partial);

  reduce_kernel<<<(NPTS * DDIM + 255) / 256, 256, 0, stream>>>(partial, out);
}